// QuantumLayer_67190468379266
// MI455X (gfx1250) — compile-verified
//
#include <hip/hip_runtime.h>
#include <math.h>

typedef __attribute__((ext_vector_type(2))) float v2f;
typedef __attribute__((ext_vector_type(8))) float v8f;

#define NQ 12
#define NSTATE 4096
#define NPADDED (4096 + 256)
#define PAD_IDX(i) ((i) + ((i) >> 4))

// CNOT ring: reference gate j = CNOT(ctrl qubit j, tgt qubit (j+1)%12).
// Reference qubit q lives at bit (11-q). Gate j therefore acts on bits
// c_j = 11-j, t_j = 11-((j+1)%12). Each gate is the involution
// sigma_j(i) = i ^ (bit_c(i) << t). After applying gates j=0..11 in order,
// s_fin[i] = s0[sigma_0(sigma_1(...sigma_11(i)))].
__device__ __forceinline__ int perm_fwd(int i) {   // tau
#pragma unroll
  for (int j = NQ - 1; j >= 0; --j) {
    int c = 11 - j;
    int t = 11 - ((j + 1) % NQ);
    i ^= ((i >> c) & 1) << t;
  }
  return i;
}
__device__ __forceinline__ int perm_inv(int i) {   // tau^{-1}
#pragma unroll
  for (int j = 0; j < NQ; ++j) {
    int c = 11 - j;
    int t = 11 - ((j + 1) % NQ);
    i ^= ((i >> c) & 1) << t;
  }
  return i;
}

// Sweep s places amplitude-index bits [4s .. 4s+3] as the "local" nibble v,
// remaining 8 bits as the group id g.
__device__ __forceinline__ int sweep_idx(int s, int g, int v) {
  if (s == 0) return (g << 4) | v;
  if (s == 1) return ((g & 0xF0) << 4) | (v << 4) | (g & 0x0F);
  return (v << 8) | g;
}

__global__ __launch_bounds__(256) void qsim_kernel(const float* __restrict__ x,
                                                   const float* __restrict__ w,
                                                   float* __restrict__ out) {
  __shared__ float sre[NPADDED];
  __shared__ float sim[NPADDED];
  __shared__ float matP[16 * 17];   // Re(U)
  __shared__ float matQ[16 * 17];   // Im(U)
  __shared__ float matQn[16 * 17];  // -Im(U)
  __shared__ float csA[NQ], snA[NQ], csB[NQ], snB[NQ];  // indexed by BIT
  __shared__ float red[8][NQ];

  const int tid = threadIdx.x;
  const int b = blockIdx.x;
  const int lane = tid & 31;
  const int wv = tid >> 5;

  // ---- per-bit angles: bit p <-> reference qubit (11-p) ----
  if (tid < NQ) {
    int q = 11 - tid;
    float thA = x[b * NQ + q] + w[q];  // fused encoding + layer-0 RX
    float thB = w[NQ + q];             // layer-1 RX
    csA[tid] = cosf(0.5f * thA);
    snA[tid] = sinf(0.5f * thA);
    csB[tid] = cosf(0.5f * thB);
    snB[tid] = sinf(0.5f * thB);
  }
  __syncthreads();

  // ---- build post-fused-layer product state directly ----
  // amp(idx) = (-i)^popcount(idx) * prod_bits (bit ? sin : cos)
  {
    float base = 1.0f;
#pragma unroll
    for (int k = 0; k < 8; ++k)
      base *= ((tid >> k) & 1) ? snA[4 + k] : csA[4 + k];
    int pct = __popc(tid);
#pragma unroll
    for (int v = 0; v < 16; ++v) {
      float m = base;
#pragma unroll
      for (int k = 0; k < 4; ++k)
        m *= ((v >> k) & 1) ? snA[k] : csA[k];
      int pc = (pct + __popc(v)) & 3;
      float re = (pc == 0) ? m : ((pc == 2) ? -m : 0.0f);
      float im = (pc == 1) ? -m : ((pc == 3) ? m : 0.0f);
      int idx = (tid << 4) | v;
      sre[PAD_IDX(idx)] = re;
      sim[PAD_IDX(idx)] = im;
    }
  }
  __syncthreads();

  // ---- CNOT ring #1: in-place gather permutation ----
  {
    float tr[16], ti[16];
#pragma unroll
    for (int v = 0; v < 16; ++v) {
      int j = perm_fwd((tid << 4) | v);
      tr[v] = sre[PAD_IDX(j)];
      ti[v] = sim[PAD_IDX(j)];
    }
    __syncthreads();
#pragma unroll
    for (int v = 0; v < 16; ++v) {
      int i = (tid << 4) | v;
      sre[PAD_IDX(i)] = tr[v];
      sim[PAD_IDX(i)] = ti[v];
    }
  }
  __syncthreads();

  // ---- layer-1 RX: 3 nibble sweeps, each a 16x16 complex matmul on WMMA ----
#pragma unroll
  for (int s = 0; s < 3; ++s) {
    // Build U = RX(b3) (x) RX(b2) (x) RX(b1) (x) RX(b0) for bits 4s..4s+3.
    // U[r][c] = m * (-i)^d, d = popcount(r^c), m = prod(bit? sin : cos).
    {
      int r = tid >> 4, c = tid & 15, xr = r ^ c;
      float m = 1.0f;
#pragma unroll
      for (int k = 0; k < 4; ++k)
        m *= ((xr >> k) & 1) ? snB[4 * s + k] : csB[4 * s + k];
      int d = __popc(xr) & 3;
      float pe = (d == 0) ? m : ((d == 2) ? -m : 0.0f);
      float qe = (d == 1) ? -m : ((d == 3) ? m : 0.0f);
      matP[r * 17 + c] = pe;
      matQ[r * 17 + c] = qe;
      matQn[r * 17 + c] = -qe;
    }
    __syncthreads();

    // A-operand (16x4 f32): lanes 0-15 -> M rows, K0/K1; lanes 16-31 -> K2/K3
    const int arow = lane & 15;
    const int koff = (lane >> 4) << 1;
    v2f aP[4], aQ[4], aQn[4];
#pragma unroll
    for (int j = 0; j < 4; ++j) {
      int k0 = 4 * j + koff;
      aP[j].x = matP[arow * 17 + k0];
      aP[j].y = matP[arow * 17 + k0 + 1];
      aQ[j].x = matQ[arow * 17 + k0];
      aQ[j].y = matQ[arow * 17 + k0 + 1];
      aQn[j].x = matQn[arow * 17 + k0];
      aQn[j].y = matQn[arow * 17 + k0 + 1];
    }

    // 256 groups -> 16 column tiles; 8 waves x 2 tiles each (disjoint, no
    // intra-sweep barrier needed: each wave reads then rewrites its own tiles)
#pragma unroll
    for (int tt = 0; tt < 2; ++tt) {
      int gcol = (wv * 2 + tt) * 16 + (lane & 15);
      // B-operand (4x16 f32): lanes 0-15 -> N cols with K0/K1; 16-31 -> K2/K3
      v2f bre[4], bim[4];
#pragma unroll
      for (int j = 0; j < 4; ++j) {
        int v0 = 4 * j + koff;
        int i0 = sweep_idx(s, gcol, v0);
        int i1 = sweep_idx(s, gcol, v0 + 1);
        bre[j].x = sre[PAD_IDX(i0)];
        bre[j].y = sre[PAD_IDX(i1)];
        bim[j].x = sim[PAD_IDX(i0)];
        bim[j].y = sim[PAD_IDX(i1)];
      }
      // re' = P@re + (-Q)@im ; im' = Q@re + P@im  (K=16 via 4 chained K=4)
      v8f dre = {};
      v8f dim = {};
#pragma unroll
      for (int j = 0; j < 4; ++j) {
        dre = __builtin_amdgcn_wmma_f32_16x16x4_f32(false, aP[j], false, bre[j],
                                                    (short)0, dre, false, false);
        dre = __builtin_amdgcn_wmma_f32_16x16x4_f32(false, aQn[j], false, bim[j],
                                                    (short)0, dre, false, false);
        dim = __builtin_amdgcn_wmma_f32_16x16x4_f32(false, aQ[j], false, bre[j],
                                                    (short)0, dim, false, false);
        dim = __builtin_amdgcn_wmma_f32_16x16x4_f32(false, aP[j], false, bim[j],
                                                    (short)0, dim, false, false);
      }
      // D layout: VGPR r, lanes 0-15 -> M=r; lanes 16-31 -> M=r+8
      int vadd = (lane >> 4) << 3;
#pragma unroll
      for (int r = 0; r < 8; ++r) {
        int idx = sweep_idx(s, gcol, r + vadd);
        sre[PAD_IDX(idx)] = dre[r];
        sim[PAD_IDX(idx)] = dim[r];
      }
    }
    __syncthreads();
  }

  // ---- probs + CNOT ring #2 fused via tau^{-1} + per-qubit <Z> ----
  float acc[NQ];
#pragma unroll
  for (int q = 0; q < NQ; ++q) acc[q] = 0.0f;
#pragma unroll
  for (int v = 0; v < 16; ++v) {
    int jdx = (tid << 4) | v;
    float re = sre[PAD_IDX(jdx)];
    float im = sim[PAD_IDX(jdx)];
    float p = re * re + im * im;
    int i = perm_inv(jdx);  // index this prob lands on after ring #2
#pragma unroll
    for (int q = 0; q < NQ; ++q) acc[q] += ((i >> q) & 1) ? -p : p;
  }
#pragma unroll
  for (int off = 16; off > 0; off >>= 1) {
#pragma unroll
    for (int q = 0; q < NQ; ++q) acc[q] += __shfl_down(acc[q], off, 32);
  }
  if (lane == 0) {
#pragma unroll
    for (int q = 0; q < NQ; ++q) red[wv][q] = acc[q];
  }
  __syncthreads();
  if (tid < NQ) {
    float sum = 0.0f;
#pragma unroll
    for (int k = 0; k < 8; ++k) sum += red[k][tid];
    out[b * NQ + (11 - tid)] = sum;  // bit tid <-> reference qubit 11-tid
  }
}

extern "C" void kernel_launch(void* const* d_in, const int* in_sizes, int n_in,
                              void* d_out, int out_size, void* d_ws,
                              size_t ws_size, hipStream_t stream) {
  const float* x = (const float*)d_in[0];
  const float* w = (const float*)d_in[1];
  float* out = (float*)d_out;
  int B = in_sizes[0] / NQ;  // 2048
  qsim_kernel<<<B, 256, 0, stream>>>(x, w, out);
}